// Node_feature_information_aggregation_36404142801484
// MI455X (gfx1250) — compile-verified
//
#include <hip/hip_runtime.h>

typedef __attribute__((ext_vector_type(2))) float v2f;
typedef __attribute__((ext_vector_type(8))) float v8f;

#define HID 128
#define NSLOPE 0.2f

// ---------------- ordered-uint mapping for float atomic max ----------------
__device__ __forceinline__ unsigned f2ord(float f) {
    int b = __float_as_int(f);
    return b < 0 ? ~(unsigned)b : ((unsigned)b | 0x80000000u);
}
__device__ __forceinline__ float ord2f(unsigned u) {
    int b = (u & 0x80000000u) ? (int)(u & 0x7fffffffu) : ~(int)u;
    return __int_as_float(b);
}

// ---------------- fused projection GEMM + attention score ----------------
// H[M x 128] = X[M x K] @ W[K x 128]           (v_wmma_f32_16x16x4_f32)
// a_out[M]   = H @ avec
// 256 threads = 8 waves; block covers 16 rows; wave w covers cols [16w,16w+16)
__global__ __launch_bounds__(256) void gemm_ha_kernel(
    const float* __restrict__ X, const float* __restrict__ W,
    const float* __restrict__ avec, float* __restrict__ H,
    float* __restrict__ a_out, int M, int K)
{
    __shared__ float As[16 * 308];   // K <= 300, padded stride K+4
    __shared__ float lds_a[16];
    const int tid  = threadIdx.x;
    const int lane = tid & 31;
    const int wv   = tid >> 5;
    const int row0 = blockIdx.x * 16;
    const int LDA  = K + 4;

    if (tid < 16) lds_a[tid] = 0.f;
    for (int idx = tid; idx < 16 * K; idx += 256) {
        int r = idx / K, c = idx - r * K;
        int gr = row0 + r;
        As[r * LDA + c] = (gr < M) ? X[(size_t)gr * K + c] : 0.f;
    }
    __syncthreads();

    const int m  = lane & 15;
    const int kk = (lane >> 4) << 1;            // 0 or 2 (A/B VGPR layout)
    const int n  = (wv << 4) + (lane & 15);

    v8f acc = {};
    for (int k0 = 0; k0 < K; k0 += 4) {         // K is always a multiple of 4
        v2f af, bf;
        af[0] = As[m * LDA + k0 + kk];
        af[1] = As[m * LDA + k0 + kk + 1];
        bf[0] = W[(size_t)(k0 + kk)     * HID + n];
        bf[1] = W[(size_t)(k0 + kk + 1) * HID + n];
        acc = __builtin_amdgcn_wmma_f32_16x16x4_f32(
                  false, af, false, bf, (short)0, acc, false, false);
    }

    const float an = avec[n];
#pragma unroll
    for (int r = 0; r < 8; ++r) {
        int ml = r + ((lane >> 4) << 3);        // C/D layout: M = r (+8 for hi lanes)
        int gr = row0 + ml;
        if (gr < M) {
            H[(size_t)gr * HID + n] = acc[r];
            atomicAdd(&lds_a[ml], acc[r] * an);
        }
    }
    __syncthreads();
    if (tid < 16 && row0 + tid < M) a_out[row0 + tid] = lds_a[tid];
}

// ---------------- wv[k] = sum_c Wd[k][c] * ad[c]  (Wd @ ad) ----------------
__global__ void wvec_kernel(const float* __restrict__ Wd,
                            const float* __restrict__ ad,
                            float* __restrict__ wv, int K)
{
    int k = blockIdx.x * blockDim.x + threadIdx.x;
    if (k >= K) return;
    float s = 0.f;
#pragma unroll 4
    for (int c = 0; c < HID; ++c) s += Wd[(size_t)k * HID + c] * ad[c];
    wv[k] = s;
}

// ---------------- y[row] = X[row] . wv   (wave per row, wave32) ----------------
__global__ __launch_bounds__(256) void matvec_kernel(
    const float* __restrict__ X, const float* __restrict__ wv,
    float* __restrict__ y, int M, int K)
{
    int row  = (blockIdx.x * blockDim.x + threadIdx.x) >> 5;
    int lane = threadIdx.x & 31;
    if (row >= M) return;
    float s = 0.f;
    for (int c = lane; c < K; c += 32) s += X[(size_t)row * K + c] * wv[c];
#pragma unroll
    for (int off = 16; off; off >>= 1) s += __shfl_xor(s, off, 32);
    if (lane == 0) y[row] = s;
}

// ---------------- init: out = bias, amax = -inf(ord), denom = 0 ----------------
__global__ void init_kernel(float* __restrict__ out, const float* __restrict__ b,
                            unsigned* __restrict__ amax, float* __restrict__ denom,
                            int n_dst)
{
    int i = blockIdx.x * blockDim.x + threadIdx.x;
    if (i < n_dst * HID) out[i] = b[i & (HID - 1)];
    if (i < n_dst) { amax[i] = 0u; denom[i] = 0.f; }
}

// ---------------- pass 1: alpha + segment max ----------------
__global__ void alpha_kernel(const int* __restrict__ edges, int E, int L,
                             const float* __restrict__ a_s,
                             const float* __restrict__ a_d,
                             float* __restrict__ alpha,
                             unsigned* __restrict__ amax)
{
    int t = blockIdx.x * blockDim.x + threadIdx.x;
    if (t >= E + L) return;
    int src, dst;
    if (t < E) { src = edges[t]; dst = edges[E + t]; }
    else       { src = t - E;    dst = t - E; }
    float x  = a_s[src] + a_d[dst];
    float al = x > 0.f ? x : NSLOPE * x;
    alpha[t] = al;
    atomicMax(&amax[dst], f2ord(al));
}

// ---------------- pass 2: ex = exp(alpha - max), segment sum ----------------
__global__ void ex_kernel(const int* __restrict__ edges, int E, int L,
                          float* __restrict__ alpha,
                          const unsigned* __restrict__ amax,
                          float* __restrict__ denom)
{
    int t = blockIdx.x * blockDim.x + threadIdx.x;
    if (t >= E + L) return;
    int dst = (t < E) ? edges[E + t] : t - E;
    float e = __expf(alpha[t] - ord2f(amax[dst]));
    alpha[t] = e;                       // reuse buffer for ex
    atomicAdd(&denom[dst], e);
}

// ---------------- pass 3: out[dst] += (ex/denom[dst]) * hs[src] ----------------
// one wave per edge; lane handles 4 channels (float4 gather, 4 f32 atomics)
__global__ __launch_bounds__(256) void scatter_kernel(
    const int* __restrict__ edges, int E, int L,
    const float* __restrict__ ex, const float* __restrict__ denom,
    const float* __restrict__ hs, float* __restrict__ out)
{
    int t    = (blockIdx.x * blockDim.x + threadIdx.x) >> 5;
    int lane = threadIdx.x & 31;
    if (t >= E + L) return;
    int src, dst;
    if (t < E) { src = edges[t]; dst = edges[E + t]; }
    else       { src = t - E;    dst = t - E; }
    float w = ex[t] / denom[dst];
    const float4 v = ((const float4*)(hs + (size_t)src * HID))[lane];
    float* orow = out + (size_t)dst * HID + lane * 4;
    atomicAdd(orow + 0, w * v.x);
    atomicAdd(orow + 1, w * v.y);
    atomicAdd(orow + 2, w * v.z);
    atomicAdd(orow + 3, w * v.w);
}

// ---------------- host-side conv driver ----------------
struct GatP { const float *Ws, *Wd, *as_, *ad_, *b; };

static void run_conv(const float* x_src, int n_src, int d_src,
                     const float* x_dst, int n_dst, int d_dst,
                     const int* edges, int E, const GatP& p,
                     float* hs, float* a_s, float* a_d, unsigned* amax,
                     float* denom, float* alpha, float* wdad, float* out,
                     hipStream_t stream)
{
    int L = n_src < n_dst ? n_src : n_dst;
    int T = E + L;
    gemm_ha_kernel<<<(n_src + 15) / 16, 256, 0, stream>>>(
        x_src, p.Ws, p.as_, hs, a_s, n_src, d_src);
    wvec_kernel<<<(d_dst + 127) / 128, 128, 0, stream>>>(p.Wd, p.ad_, wdad, d_dst);
    matvec_kernel<<<(n_dst * 32 + 255) / 256, 256, 0, stream>>>(
        x_dst, wdad, a_d, n_dst, d_dst);
    init_kernel<<<((long long)n_dst * HID + 255) / 256, 256, 0, stream>>>(
        out, p.b, amax, denom, n_dst);
    alpha_kernel<<<(T + 255) / 256, 256, 0, stream>>>(edges, E, L, a_s, a_d, alpha, amax);
    ex_kernel<<<(T + 255) / 256, 256, 0, stream>>>(edges, E, L, alpha, amax, denom);
    scatter_kernel<<<((long long)T * 32 + 255) / 256, 256, 0, stream>>>(
        edges, E, L, alpha, denom, hs, out);
}

extern "C" void kernel_launch(void* const* d_in, const int* in_sizes, int n_in,
                              void* d_out, int out_size, void* d_ws, size_t ws_size,
                              hipStream_t stream)
{
    (void)n_in; (void)out_size; (void)ws_size;
    const int N = 50000, DL = 256, DP = 300, E = 800000;

    const float* lnc_x  = (const float*)d_in[0];
    const float* prot_x = (const float*)d_in[1];
    const int*   e_lp   = (const int*)d_in[2];
    const int*   e_pl   = (const int*)d_in[3];

    // params: 4 groups (l1.lp, l1.pl, l2.lp, l2.pl) of 5 arrays from d_in[4].
    // jax pytree flatten sorts dict keys -> (Wd, Ws, ad, as, b); detect via sizes.
    bool wd_first = (in_sizes[4] == DP * HID);   // l1.lp: Wd=300x128, Ws=256x128
    GatP P[4];
    for (int g = 0; g < 4; ++g) {
        const float* q0 = (const float*)d_in[4 + 5 * g + 0];
        const float* q1 = (const float*)d_in[4 + 5 * g + 1];
        const float* q2 = (const float*)d_in[4 + 5 * g + 2];
        const float* q3 = (const float*)d_in[4 + 5 * g + 3];
        const float* q4 = (const float*)d_in[4 + 5 * g + 4];
        if (wd_first) { P[g].Wd = q0; P[g].Ws = q1; P[g].ad_ = q2; P[g].as_ = q3; }
        else          { P[g].Ws = q0; P[g].Wd = q1; P[g].as_ = q2; P[g].ad_ = q3; }
        P[g].b = q4;
    }

    // workspace layout (floats)
    float* ws = (float*)d_ws;
    size_t o = 0;
    float*    lnc1  = ws + o; o += (size_t)N * HID;
    float*    prot1 = ws + o; o += (size_t)N * HID;
    float*    hs    = ws + o; o += (size_t)N * HID;
    float*    a_s   = ws + o; o += N;
    float*    a_d   = ws + o; o += N;
    unsigned* amax  = (unsigned*)(ws + o); o += N;
    float*    denom = ws + o; o += N;
    float*    alpha = ws + o; o += (size_t)E + N;
    float*    wdad  = ws + o; o += 512;

    float* lnc_out  = (float*)d_out;                       // returned first
    float* prot_out = (float*)d_out + (size_t)N * HID;     // returned second

    // layer 1 (both convs read the ORIGINAL features)
    run_conv(lnc_x,  N, DL, prot_x, N, DP, e_lp, E, P[0],
             hs, a_s, a_d, amax, denom, alpha, wdad, prot1, stream);
    run_conv(prot_x, N, DP, lnc_x,  N, DL, e_pl, E, P[1],
             hs, a_s, a_d, amax, denom, alpha, wdad, lnc1, stream);
    // layer 2 (reads layer-1 outputs), writes straight into d_out
    run_conv(lnc1,  N, HID, prot1, N, HID, e_lp, E, P[2],
             hs, a_s, a_d, amax, denom, alpha, wdad, prot_out, stream);
    run_conv(prot1, N, HID, lnc1,  N, HID, e_pl, E, P[3],
             hs, a_s, a_d, amax, denom, alpha, wdad, lnc_out, stream);
}